// SemanticSegmentationClassifier_39831526703159
// MI455X (gfx1250) — compile-verified
//
#include <hip/hip_runtime.h>
#include <hip/hip_bf16.h>

#define NB   2
#define CB   256
#define HH   96
#define WW   96
#define HWP  (HH*WW)          // 9216
#define NPOS (NB*HWP)         // 18432
#define HO   384
#define WO   384
#define TT   256
#define NC1  3
#define NC2  5

// conv tiling
#define TS    8               // spatial tile 8x8
#define PS    10              // halo patch 10x10
#define NPIX  (PS*PS)         // 100
#define ASTR  146             // As row stride (144 + pad, even)
#define PSTR  18              // Ps row stride (16 + pad, even)
#define NTX   (WW/TS)         // 12
#define NTY   (HH/TS)         // 12
#define PTILES (NB*NTY*NTX)   // 288
#define NCHUNK (CB/16)        // 16

typedef __attribute__((ext_vector_type(2))) float v2f;
typedef __attribute__((ext_vector_type(8))) float v8f;

#if __has_builtin(__builtin_amdgcn_global_load_async_to_lds_b32)
#define ASYNC_STAGE 1
#else
#define ASYNC_STAGE 0
#endif

#if ASYNC_STAGE
typedef __attribute__((address_space(1))) int g_i32;
typedef __attribute__((address_space(3))) int l_i32;

__device__ __forceinline__ void async_ld_b32(const float* g, float* l)
{
    __builtin_amdgcn_global_load_async_to_lds_b32(
        (g_i32*)(g), (l_i32*)(l), 0, 0);
}
__device__ __forceinline__ void wait_async_25()
{
#if __has_builtin(__builtin_amdgcn_s_wait_asynccnt)
    __builtin_amdgcn_s_wait_asynccnt(25);
#else
    asm volatile("s_wait_asynccnt 0x19" ::: "memory");
#endif
}
__device__ __forceinline__ void wait_async_0()
{
#if __has_builtin(__builtin_amdgcn_s_wait_asynccnt)
    __builtin_amdgcn_s_wait_asynccnt(0);
#else
    asm volatile("s_wait_asynccnt 0x0" ::: "memory");
#endif
}
#endif

// ---------------------------------------------------------------------------
// 3x3 SAME conv, implicit GEMM on the f32 WMMA pipe, 2D tiles with halo.
// Block = 256 threads = 8 waves (2 cout-waves x 4 pos-waves).
// Tile  = 32 cout x 64 positions (8x8 spatial).
// Per 16-ci chunk: 10x10 halo + 9-tap weights staged via async-to-LDS
// (double buffered) -> 36 WMMAs per chunk on the other buffer.
// ---------------------------------------------------------------------------
__global__ __launch_bounds__(256)
void sseg_conv3x3_wmma(const float* __restrict__ in,
                       const float* __restrict__ w,
                       float* __restrict__ out)
{
#if ASYNC_STAGE
    __shared__ float As[2][32 * ASTR];   // [buf][cout][tap*16+ci]
    __shared__ float Ps[2][NPIX * PSTR]; // [buf][pix][ci]
    __shared__ float Trash[64];          // sink for OOB halo lanes
#else
    __shared__ float As[1][32 * ASTR];
    __shared__ float Ps[1][NPIX * PSTR];
#endif

    const int tid   = threadIdx.x;
    const int lane  = tid & 31;
    const int wid   = tid >> 5;
    const int half  = lane >> 4;      // lanes 16-31 hold K+2 (ISA A/B layout)
    const int l15   = lane & 15;
    const int waveM = wid >> 2;       // 0..1 -> cout sub-tile
    const int waveN = wid & 3;        // 0..3 -> pos sub-tile

    const int pt     = blockIdx.x % PTILES;
    const int coTile = blockIdx.x / PTILES;
    const int coBase = coTile * 32;
    const int b      = pt / (NTY * NTX);
    const int tr     = pt - b * (NTY * NTX);
    const int Y0     = (tr / NTX) * TS;
    const int X0     = (tr - (tr / NTX) * NTX) * TS;

    // ---- hoisted staging descriptors -------------------------------------
    // A: 32 cout x 144 (ci*9+tap contiguous in OIHW) -> 18 slots/thread
    int aDst[18], aSrc[18];
#pragma unroll
    for (int s = 0; s < 18; ++s) {
        int idx = s * 256 + tid;            // 0..4607
        int co  = idx / 144;
        int r   = idx - co * 144;           // r = ci*9 + tap
        int ci  = r / 9;
        int tap = r - ci * 9;
        aDst[s] = co * ASTR + tap * 16 + ci;
        aSrc[s] = (coBase + co) * (CB * 9) + r;   // + ci0*9 added per chunk
    }
    // P: 16 ci x 100 pix -> 7 slots/thread
    int pDst[7], pOff[7], pCi[7];
    bool pInb[7], pValid[7];
#pragma unroll
    for (int s = 0; s < 7; ++s) {
        int idx = s * 256 + tid;            // 0..1791, valid < 1600
        pValid[s] = idx < 1600;
        int ci = idx / NPIX;
        int p  = idx - ci * NPIX;
        int py = p / PS, px = p - py * PS;
        int gy = Y0 - 1 + py, gx = X0 - 1 + px;
        pInb[s] = pValid[s] && gy >= 0 && gy < HH && gx >= 0 && gx < WW;
        pOff[s] = gy * WW + gx;
        pCi[s]  = ci;
        pDst[s] = p * PSTR + ci;
    }
    const int inBase = b * CB * HWP;

    // per-lane read bases for the MMA loop
    const int nCol  = waveN * 16 + l15;             // 0..63 position in tile
    const int pyL   = nCol >> 3, pxL = nCol & 7;
    const int pBase = ((pyL + 1) * PS + (pxL + 1)) * PSTR + 2 * half;
    const int aRow  = (waveM * 16 + l15) * ASTR + 2 * half;

    v8f acc = {};

#if ASYNC_STAGE
    // zero the never-loaded (OOB halo) patch slots in both buffers, once
#pragma unroll
    for (int s = 0; s < 7; ++s)
        if (pValid[s] && !pInb[s]) { Ps[0][pDst[s]] = 0.f; Ps[1][pDst[s]] = 0.f; }

    // prologue: issue chunk 0 into buffer 0 (25 async ops per thread)
    {
#pragma unroll
        for (int s = 0; s < 18; ++s)
            async_ld_b32(&w[aSrc[s]], &As[0][aDst[s]]);
#pragma unroll
        for (int s = 0; s < 7; ++s) {
            const float* g = pInb[s] ? &in[inBase + pCi[s] * HWP + pOff[s]] : in;
            float*       l = pInb[s] ? &Ps[0][pDst[s]] : &Trash[lane];
            async_ld_b32(g, l);
        }
    }

    for (int i = 0; i < NCHUNK; ++i) {
        const int cur = i & 1;
        if (i + 1 < NCHUNK) {
            const int nxt  = cur ^ 1;
            const int aAdd = (i + 1) * 16 * 9;
            const int cAdd = inBase + (i + 1) * 16 * HWP;
#pragma unroll
            for (int s = 0; s < 18; ++s)
                async_ld_b32(&w[aSrc[s] + aAdd], &As[nxt][aDst[s]]);
#pragma unroll
            for (int s = 0; s < 7; ++s) {
                const float* g = pInb[s] ? &in[cAdd + pCi[s] * HWP + pOff[s]] : in;
                float*       l = pInb[s] ? &Ps[nxt][pDst[s]] : &Trash[lane];
                async_ld_b32(g, l);
            }
            wait_async_25();      // chunk i complete, chunk i+1 in flight
        } else {
            wait_async_0();
        }
        __syncthreads();          // chunk i visible to all waves

        const float* AsB = &As[cur][0];
        const float* PsB = &Ps[cur][0];
#pragma unroll
        for (int tap = 0; tap < 9; ++tap) {
            const int dy = tap / 3 - 1, dx = tap % 3 - 1;
            const int pIdx = pBase + (dy * PS + dx) * PSTR;
            const int aIdx = aRow + tap * 16;
#pragma unroll
            for (int kk = 0; kk < 16; kk += 4) {
                v2f a  = *(const v2f*)&AsB[aIdx + kk];
                v2f bb = *(const v2f*)&PsB[pIdx + kk];
                acc = __builtin_amdgcn_wmma_f32_16x16x4_f32(
                        false, a, false, bb, (short)0, acc, false, false);
            }
        }
        __syncthreads();          // done reading buf[cur] before it is refilled
    }
#else
    for (int ci0 = 0; ci0 < CB; ci0 += 16) {
        __syncthreads();
        const int aAdd = ci0 * 9;
#pragma unroll
        for (int s = 0; s < 18; ++s)
            As[0][aDst[s]] = w[aSrc[s] + aAdd];
        const int cAdd = inBase + ci0 * HWP;
#pragma unroll
        for (int s = 0; s < 7; ++s) {
            float v = 0.f;
            if (pInb[s]) v = in[cAdd + pCi[s] * HWP + pOff[s]];
            if (pValid[s]) Ps[0][pDst[s]] = v;
        }
        __syncthreads();
#pragma unroll
        for (int tap = 0; tap < 9; ++tap) {
            const int dy = tap / 3 - 1, dx = tap % 3 - 1;
            const int pIdx = pBase + (dy * PS + dx) * PSTR;
            const int aIdx = aRow + tap * 16;
#pragma unroll
            for (int kk = 0; kk < 16; kk += 4) {
                v2f a  = *(const v2f*)&As[0][aIdx + kk];
                v2f bb = *(const v2f*)&Ps[0][pIdx + kk];
                acc = __builtin_amdgcn_wmma_f32_16x16x4_f32(
                        false, a, false, bb, (short)0, acc, false, false);
            }
        }
    }
#endif

    // store 16x16 tile: lane owns position nCol; row r -> cout (r + 8*half)
    {
        const int sp = (Y0 + pyL) * WW + (X0 + pxL);
#pragma unroll
        for (int r = 0; r < 8; ++r) {
            int co = coBase + waveM * 16 + r + 8 * half;
            out[inBase + co * HWP + sp] = acc[r];
        }
    }
}

// ---------------------------------------------------------------------------
// Per-channel batch stats (mean, rsqrt(var+eps)); one block per channel.
// ---------------------------------------------------------------------------
__global__ __launch_bounds__(256)
void sseg_bn_stats(const float* __restrict__ x, float* __restrict__ mv)
{
    const int c = blockIdx.x;
    float s = 0.f, s2 = 0.f;
    for (int b = 0; b < NB; ++b) {
        const float* p = x + (b * CB + c) * HWP;
        for (int i = threadIdx.x; i < HWP; i += 256) {
            float v = p[i]; s += v; s2 += v * v;
        }
    }
    __shared__ float sh[256], sh2[256];
    sh[threadIdx.x] = s; sh2[threadIdx.x] = s2;
    __syncthreads();
    for (int o = 128; o > 0; o >>= 1) {
        if (threadIdx.x < o) {
            sh[threadIdx.x]  += sh[threadIdx.x + o];
            sh2[threadIdx.x] += sh2[threadIdx.x + o];
        }
        __syncthreads();
    }
    if (threadIdx.x == 0) {
        float inv_n = 1.f / (float)(NB * HWP);
        float m = sh[0] * inv_n;
        float v = sh2[0] * inv_n - m * m;
        mv[2 * c]     = m;
        mv[2 * c + 1] = rsqrtf(v + 1e-5f);
    }
}

__global__ __launch_bounds__(256)
void sseg_bn_relu(const float* __restrict__ x, const float* __restrict__ mv,
                  const float* __restrict__ g, const float* __restrict__ bta,
                  float* __restrict__ out)
{
    int i = blockIdx.x * 256 + threadIdx.x;   // grid sized exactly
    int c = (i / HWP) % CB;
    float v = (x[i] - mv[2 * c]) * mv[2 * c + 1] * g[c] + bta[c];
    out[i] = fmaxf(v, 0.f);
}

// ---------------------------------------------------------------------------
// Head: 1x1 convs at 96x96 (upsample commutes with 1x1 conv), replicate 4x4
// into x1/x2, paint labels at 384^2 from the sorted boxes, accumulate CE.
// ---------------------------------------------------------------------------
__global__ __launch_bounds__(128)
void sseg_head(const float* __restrict__ act,
               const float* __restrict__ w31, const float* __restrict__ b31,
               const float* __restrict__ w32, const float* __restrict__ b32,
               const int* __restrict__ seg_idx, const int* __restrict__ seg_cls,
               const int* __restrict__ coors,
               float* __restrict__ x1, float* __restrict__ x2,
               float* __restrict__ partial)
{
    __shared__ float sw[(NC1 + NC2) * CB];
    for (int i = threadIdx.x; i < NC1 * CB; i += 128) sw[i] = w31[i];
    for (int i = threadIdx.x; i < NC2 * CB; i += 128) sw[NC1 * CB + i] = w32[i];
    __syncthreads();

    const int t = blockIdx.x * 128 + threadIdx.x;   // 0..NPOS-1 exact grid
    float loss = 0.f;

    const int b   = t / HWP;
    const int rem = t - b * HWP;
    const int y   = rem / WW;
    const int x   = rem - y * WW;

    float z1[NC1] = {}, z2[NC2] = {};
    const float* pa = act + b * CB * HWP + rem;
    for (int c = 0; c < CB; ++c) {
        float v = pa[c * HWP];
#pragma unroll
        for (int o = 0; o < NC1; ++o) z1[o] += v * sw[o * CB + c];
#pragma unroll
        for (int o = 0; o < NC2; ++o) z2[o] += v * sw[NC1 * CB + o * CB + c];
    }
#pragma unroll
    for (int o = 0; o < NC1; ++o) z1[o] += b31[o];
#pragma unroll
    for (int o = 0; o < NC2; ++o) z2[o] += b32[o];

    // replicate logits into the 4x4 up-sampled outputs
#pragma unroll
    for (int o = 0; o < NC1; ++o) {
        int base = ((b * NC1 + o) * HO + 4 * y) * WO + 4 * x;
#pragma unroll
        for (int sy = 0; sy < 4; ++sy)
#pragma unroll
            for (int sx = 0; sx < 4; ++sx)
                x1[base + sy * WO + sx] = z1[o];
    }
#pragma unroll
    for (int o = 0; o < NC2; ++o) {
        int base = ((b * NC2 + o) * HO + 4 * y) * WO + 4 * x;
#pragma unroll
        for (int sy = 0; sy < 4; ++sy)
#pragma unroll
            for (int sx = 0; sx < 4; ++sx)
                x2[base + sy * WO + sx] = z2[o];
    }

    // stable log-softmax
    float m1 = fmaxf(fmaxf(z1[0], z1[1]), z1[2]);
    float e1 = 0.f;
#pragma unroll
    for (int o = 0; o < NC1; ++o) e1 += __expf(z1[o] - m1);
    float l1 = m1 + __logf(e1);
    float lp1[NC1];
#pragma unroll
    for (int o = 0; o < NC1; ++o) lp1[o] = z1[o] - l1;

    float m2 = z2[0];
#pragma unroll
    for (int o = 1; o < NC2; ++o) m2 = fmaxf(m2, z2[o]);
    float e2 = 0.f;
#pragma unroll
    for (int o = 0; o < NC2; ++o) e2 += __expf(z2[o] - m2);
    float l2 = m2 + __logf(e2);
    float lp2[NC2];
#pragma unroll
    for (int o = 0; o < NC2; ++o) lp2[o] = z2[o] - l2;

    // paint labels for the 16 sub-pixels (sequential overwrite, seg!=prev dedup)
    int pn[16], cl[16];
#pragma unroll
    for (int i = 0; i < 16; ++i) { pn[i] = 0; cl[i] = 0; }
    const int X0 = 4 * x, Y0 = 4 * y;
    const int* sI = seg_idx + b * TT;
    const int* sC = seg_cls + b * TT;
    const int* cr = coors + b * TT * 4;
    int prev = -1;
    for (int tk = 0; tk < TT; ++tk) {
        int s = sI[tk];
        bool active = (s != prev);
        prev = s;
        if (!active) continue;
        int bx0 = cr[4 * tk + 0], by0 = cr[4 * tk + 1];
        int bx1 = cr[4 * tk + 2], by1 = cr[4 * tk + 3];
        if (bx1 <= X0 || bx0 >= X0 + 4 || by1 <= Y0 || by0 >= Y0 + 4) continue;
        int c  = sC[tk];
        int pv = (c > 0) ? 1 : 2;
#pragma unroll
        for (int sy = 0; sy < 4; ++sy)
#pragma unroll
            for (int sx = 0; sx < 4; ++sx) {
                int yy = Y0 + sy, xx = X0 + sx;
                if (yy >= by0 && yy < by1 && xx >= bx0 && xx < bx1) {
                    pn[sy * 4 + sx] = pv;
                    cl[sy * 4 + sx] = c;
                }
            }
    }
#pragma unroll
    for (int i = 0; i < 16; ++i) {
        int p = pn[i], c = cl[i];
        float a = (p == 0) ? lp1[0] : (p == 1) ? lp1[1] : lp1[2];
        float d = (c == 0) ? lp2[0] : (c == 1) ? lp2[1] :
                  (c == 2) ? lp2[2] : (c == 3) ? lp2[3] : lp2[4];
        loss -= a + d;
    }

    __shared__ float red[128];
    red[threadIdx.x] = loss;
    __syncthreads();
    for (int o = 64; o > 0; o >>= 1) {
        if (threadIdx.x < o) red[threadIdx.x] += red[threadIdx.x + o];
        __syncthreads();
    }
    if (threadIdx.x == 0) partial[blockIdx.x] = red[0];
}

__global__ void sseg_finalize(const float* __restrict__ partial, int n,
                              float* __restrict__ out)
{
    if (threadIdx.x == 0 && blockIdx.x == 0) {
        float s = 0.f;
        for (int i = 0; i < n; ++i) s += partial[i];   // fixed order: deterministic
        out[0] = s / (float)(NB * HO * WO);
    }
}

// ---------------------------------------------------------------------------
extern "C" void kernel_launch(void* const* d_in, const int* in_sizes, int n_in,
                              void* d_out, int out_size, void* d_ws, size_t ws_size,
                              hipStream_t stream)
{
    const float* fuse   = (const float*)d_in[0];
    const int*   segi   = (const int*)  d_in[1];
    const int*   segc   = (const int*)  d_in[2];
    const int*   coors  = (const int*)  d_in[3];
    const float* c1w    = (const float*)d_in[4];
    const float* bn1g   = (const float*)d_in[5];
    const float* bn1b   = (const float*)d_in[6];
    const float* c2w    = (const float*)d_in[7];
    const float* bn2g   = (const float*)d_in[8];
    const float* bn2b   = (const float*)d_in[9];
    const float* c31w   = (const float*)d_in[10];
    const float* c31b   = (const float*)d_in[11];
    const float* c32w   = (const float*)d_in[12];
    const float* c32b   = (const float*)d_in[13];

    float* out  = (float*)d_out;
    float* loss = out;
    float* x1   = out + 1;
    float* x2   = out + 1 + (size_t)NB * NC1 * HO * WO;

    const size_t FEAT = (size_t)NB * CB * HWP;       // 4,718,592 floats
    float* ws   = (float*)d_ws;
    float* buf0 = ws;                                 // raw conv out
    float* buf1 = ws + FEAT;                          // bn+relu out
    float* mv   = ws + 2 * FEAT;                      // 512 floats
    float* part = mv + 2 * CB;                        // 144 floats

    const int convBlocks = PTILES * (CB / 32);        // 2304
    const int ewBlocks   = (int)(FEAT / 256);         // 18432
    const int headBlocks = NPOS / 128;                // 144

    // stage 1: conv1 -> BN1 -> ReLU
    sseg_conv3x3_wmma<<<convBlocks, 256, 0, stream>>>(fuse, c1w, buf0);
    sseg_bn_stats   <<<CB,        256, 0, stream>>>(buf0, mv);
    sseg_bn_relu    <<<ewBlocks,  256, 0, stream>>>(buf0, mv, bn1g, bn1b, buf1);

    // stage 2: conv2 -> BN2 -> ReLU (reuse buffers)
    sseg_conv3x3_wmma<<<convBlocks, 256, 0, stream>>>(buf1, c2w, buf0);
    sseg_bn_stats   <<<CB,        256, 0, stream>>>(buf0, mv);
    sseg_bn_relu    <<<ewBlocks,  256, 0, stream>>>(buf0, mv, bn2g, bn2b, buf1);

    // stage 3: heads + labels + CE
    sseg_head<<<headBlocks, 128, 0, stream>>>(buf1, c31w, c31b, c32w, c32b,
                                              segi, segc, coors, x1, x2, part);
    sseg_finalize<<<1, 1, 0, stream>>>(part, headBlocks, loss);
}